// GNNAnomalyDetector_47699906790066
// MI455X (gfx1250) — compile-verified
//
#include <hip/hip_runtime.h>
#include <hip/hip_bf16.h>

typedef __attribute__((ext_vector_type(2))) float v2f;
typedef __attribute__((ext_vector_type(8))) float v8f;

// ---------------------------------------------------------------------------
// Degree / normalization kernels
// ---------------------------------------------------------------------------
__global__ void k_deg_init(float* deg, int N) {
    int i = blockIdx.x * blockDim.x + threadIdx.x;
    if (i < N) deg[i] = 1.0f;               // self loop contributes 1
}

__global__ void k_deg_edges(const int* __restrict__ dst, int E, float* deg) {
    int e = blockIdx.x * blockDim.x + threadIdx.x;
    if (e < E) atomicAdd(&deg[dst[e]], 1.0f);
}

__global__ void k_deg_to_dinv(float* dinv, int N) {
    int i = blockIdx.x * blockDim.x + threadIdx.x;
    if (i < N) dinv[i] = rsqrtf(dinv[i]);   // deg >= 1 always
}

// ---------------------------------------------------------------------------
// WMMA GEMM: H[N x DOUT] = X[N x DIN] @ W[DIN x DOUT], f32 exact via
// V_WMMA_F32_16X16X4_F32. One wave handles 16 rows and all DOUT columns.
// Block = 128 threads = 4 waves = 64 rows.
// ---------------------------------------------------------------------------
template <int DIN, int DOUT>
__global__ void k_gemm_wmma(const float* __restrict__ X,
                            const float* __restrict__ W,
                            float* __restrict__ H, int N) {
    constexpr int NT = (DOUT + 15) / 16;     // 16-wide column tiles
    const int wave = threadIdx.x >> 5;
    const int lane = threadIdx.x & 31;
    const int base = (blockIdx.x * (blockDim.x >> 5) + wave) * 16;
    if (base >= N) return;                   // wave-uniform: EXEC stays full

    const int half = lane >> 4;              // 0: K=0,1  1: K=2,3  (A layout)
    const int lid  = lane & 15;
    int rowA = base + lid;
    if (rowA >= N) rowA = N - 1;             // clamp; garbage rows never stored

    v8f c[NT] = {};

    for (int k = 0; k < DIN; k += 4) {
        v2f a;
        a.x = X[(long long)rowA * DIN + k + 2 * half + 0];
        a.y = X[(long long)rowA * DIN + k + 2 * half + 1];
#pragma unroll
        for (int t = 0; t < NT; ++t) {
            const int col = t * 16 + lid;
            v2f b;
            b.x = (col < DOUT) ? W[(k + 2 * half + 0) * DOUT + col] : 0.0f;
            b.y = (col < DOUT) ? W[(k + 2 * half + 1) * DOUT + col] : 0.0f;
            // D = A*B + C  (f32 16x16x4)
            c[t] = __builtin_amdgcn_wmma_f32_16x16x4_f32(
                false, a, false, b, (short)0, c[t], false, false);
        }
    }

    // C/D layout: vgpr r, lanes 0-15 -> row base+r, lanes 16-31 -> row base+8+r
#pragma unroll
    for (int t = 0; t < NT; ++t) {
        const int col = t * 16 + lid;
        if (col < DOUT) {
#pragma unroll
            for (int r = 0; r < 8; ++r) {
                const int row = base + r + 8 * half;
                if (row < N) H[(long long)row * DOUT + col] = c[t][r];
            }
        }
    }
}

// ---------------------------------------------------------------------------
// Propagation kernels
// ---------------------------------------------------------------------------
// acc[i,f] = h[i,f] * dinv[i]^2   (self-loop term; also initializes acc)
template <int D>
__global__ void k_self_init(const float* __restrict__ h,
                            const float* __restrict__ dinv,
                            float* __restrict__ acc, int N) {
    long long idx = (long long)blockIdx.x * blockDim.x + threadIdx.x;
    if (idx >= (long long)N * D) return;
    int i = (int)(idx / D);
    float di = dinv[i];
    acc[idx] = h[idx] * di * di;
}

// scatter-add: acc[dst] += h[src] * dinv[src]*dinv[dst]; 4 features/thread
template <int D>
__global__ void k_edge_prop(const int* __restrict__ src,
                            const int* __restrict__ dst,
                            const float* __restrict__ dinv,
                            const float* __restrict__ h,
                            float* __restrict__ acc, int E) {
    constexpr int G = D / 4;                 // float4 groups per edge
    long long tid = (long long)blockIdx.x * blockDim.x + threadIdx.x;
    if (tid >= (long long)E * G) return;
    int e = (int)(tid / G);
    int g = (int)(tid % G);
    int s = src[e], d = dst[e];
    float nrm = dinv[s] * dinv[d];
    const float4 hv = ((const float4*)(h + (long long)s * D))[g];
    float* ap = acc + (long long)d * D + (long long)g * 4;
    atomicAdd(ap + 0, hv.x * nrm);
    atomicAdd(ap + 1, hv.y * nrm);
    atomicAdd(ap + 2, hv.z * nrm);
    atomicAdd(ap + 3, hv.w * nrm);
}

template <int D, bool RELU>
__global__ void k_bias_act(const float* __restrict__ acc,
                           const float* __restrict__ bias,
                           float* __restrict__ out, int N) {
    long long idx = (long long)blockIdx.x * blockDim.x + threadIdx.x;
    if (idx >= (long long)N * D) return;
    int f = (int)(idx % D);
    float v = acc[idx] + bias[f];
    out[idx] = RELU ? fmaxf(v, 0.0f) : v;
}

// ---------------------------------------------------------------------------
// Launch
// ---------------------------------------------------------------------------
static inline int cdiv(long long a, long long b) { return (int)((a + b - 1) / b); }

extern "C" void kernel_launch(void* const* d_in, const int* in_sizes, int n_in,
                              void* d_out, int out_size, void* d_ws, size_t ws_size,
                              hipStream_t stream) {
    const float* x  = (const float*)d_in[0];
    const int*   ei = (const int*)d_in[1];
    const float* W1 = (const float*)d_in[2]; const float* b1 = (const float*)d_in[3];
    const float* W2 = (const float*)d_in[4]; const float* b2 = (const float*)d_in[5];
    const float* W3 = (const float*)d_in[6]; const float* b3 = (const float*)d_in[7];
    const float* W4 = (const float*)d_in[8]; const float* b4 = (const float*)d_in[9];

    const int N = in_sizes[0] / 64;          // 100000
    const int E = in_sizes[1] / 2;           // 3200000
    const int* src = ei;
    const int* dst = ei + E;

    float* dinv = (float*)d_ws;              // N floats
    float* h    = dinv + N;                  // N*64 floats (gemm output)
    float* acc  = h + (size_t)N * 64;        // N*64 floats (propagation output)
    float* out  = (float*)d_out;             // N*64 floats

    const int B = 256;

    // --- normalization ---
    k_deg_init<<<cdiv(N, B), B, 0, stream>>>(dinv, N);
    k_deg_edges<<<cdiv(E, B), B, 0, stream>>>(dst, E, dinv);
    k_deg_to_dinv<<<cdiv(N, B), B, 0, stream>>>(dinv, N);

    // --- layer 1: 64 -> 32, relu ---
    k_gemm_wmma<64, 32><<<cdiv(N, 64), 128, 0, stream>>>(x, W1, h, N);
    k_self_init<32><<<cdiv((long long)N * 32, B), B, 0, stream>>>(h, dinv, acc, N);
    k_edge_prop<32><<<cdiv((long long)E * 8, B), B, 0, stream>>>(src, dst, dinv, h, acc, E);
    k_bias_act<32, true><<<cdiv((long long)N * 32, B), B, 0, stream>>>(acc, b1, acc, N);

    // --- layer 2: 32 -> 16, relu ---
    k_gemm_wmma<32, 16><<<cdiv(N, 64), 128, 0, stream>>>(acc, W2, h, N);
    k_self_init<16><<<cdiv((long long)N * 16, B), B, 0, stream>>>(h, dinv, acc, N);
    k_edge_prop<16><<<cdiv((long long)E * 4, B), B, 0, stream>>>(src, dst, dinv, h, acc, E);
    k_bias_act<16, true><<<cdiv((long long)N * 16, B), B, 0, stream>>>(acc, b2, acc, N);

    // --- layer 3: 16 -> 8, relu ---
    k_gemm_wmma<16, 8><<<cdiv(N, 64), 128, 0, stream>>>(acc, W3, h, N);
    k_self_init<8><<<cdiv((long long)N * 8, B), B, 0, stream>>>(h, dinv, acc, N);
    k_edge_prop<8><<<cdiv((long long)E * 2, B), B, 0, stream>>>(src, dst, dinv, h, acc, E);
    k_bias_act<8, true><<<cdiv((long long)N * 8, B), B, 0, stream>>>(acc, b3, acc, N);

    // --- layer 4: 8 -> 64, no relu, write d_out ---
    k_gemm_wmma<8, 64><<<cdiv(N, 64), 128, 0, stream>>>(acc, W4, h, N);
    k_self_init<64><<<cdiv((long long)N * 64, B), B, 0, stream>>>(h, dinv, acc, N);
    k_edge_prop<64><<<cdiv((long long)E * 16, B), B, 0, stream>>>(src, dst, dinv, h, acc, E);
    k_bias_act<64, false><<<cdiv((long long)N * 64, B), B, 0, stream>>>(acc, b4, out, N);
}